// PNALayer_41807211660016
// MI455X (gfx1250) — compile-verified
//
#include <hip/hip_runtime.h>

typedef __attribute__((ext_vector_type(2))) float v2f;
typedef __attribute__((ext_vector_type(4))) float f4;
typedef __attribute__((ext_vector_type(8))) float v8f;

#define FLT_BIG 3.402823466e+38f
#define EPS_PNA 1e-5f

static __device__ __forceinline__ v8f wmma_f32(v2f a, v2f b, v8f c) {
  // V_WMMA_F32_16X16X4_F32: D = A(16x4) x B(4x16) + C(16x16), all f32
  return __builtin_amdgcn_wmma_f32_16x16x4_f32(false, a, false, b, (short)0, c,
                                               false, false);
}

static __device__ __forceinline__ void atom_add(float* p, float v) {
  __hip_atomic_fetch_add(p, v, __ATOMIC_RELAXED, __HIP_MEMORY_SCOPE_AGENT);
}
static __device__ __forceinline__ void atom_max(float* p, float v) {
  __hip_atomic_fetch_max(p, v, __ATOMIC_RELAXED, __HIP_MEMORY_SCOPE_AGENT);
}
static __device__ __forceinline__ void atom_min(float* p, float v) {
  __hip_atomic_fetch_min(p, v, __ATOMIC_RELAXED, __HIP_MEMORY_SCOPE_AGENT);
}

// ---------------- init: zero sums/cnt, +-inf max/min, zero avg accumulator --
__global__ void pna_init(float* __restrict__ sum, float* __restrict__ sumsq,
                         float* __restrict__ gmx, float* __restrict__ gmn,
                         float* __restrict__ cnt, float* __restrict__ avga,
                         long T, int n) {
  long i = (long)blockIdx.x * blockDim.x + threadIdx.x;
  if (i < T) {
    sum[i] = 0.f; sumsq[i] = 0.f; gmx[i] = -FLT_BIG; gmn[i] = FLT_BIG;
  }
  if (i < n) cnt[i] = 0.f;
  if (i == 0) *avga = 0.f;
}

// ---------------- fold edge encoder into W_pre's third block (transposed) --
// Wct[n][12] = (W_ee @ W_pre[256:384])^T   (cols 10,11 zero)
// bc[128]    = b_pre + b_ee @ W_pre[256:384]
__global__ void pna_prep(const float* __restrict__ W_ee,
                         const float* __restrict__ b_ee,
                         const float* __restrict__ W_pre,
                         const float* __restrict__ b_pre,
                         float* __restrict__ Wct, float* __restrict__ bc) {
  int n = threadIdx.x;  // 0..127
  for (int i = 0; i < 12; ++i) {
    float acc = 0.f;
    if (i < 10)
      for (int d = 0; d < 128; ++d)
        acc += W_ee[i * 128 + d] * W_pre[(256 + d) * 128 + n];
    Wct[n * 12 + i] = acc;
  }
  float bb = b_pre[n];
  for (int d = 0; d < 128; ++d) bb += b_ee[d] * W_pre[(256 + d) * 128 + n];
  bc[n] = bb;
}

// ---------------- weight transpose: dst[n*K+k] = src[(rowOff+k)*128+n] -----
// Makes each lane's B-fragment pair contiguous -> single global_load_b64.
__global__ void pna_transpose(const float* __restrict__ src,
                              float* __restrict__ dst, int K, int rowOff) {
  long i = (long)blockIdx.x * blockDim.x + threadIdx.x;
  if (i < (long)K * 128) {
    int n = (int)(i / K);
    int k = (int)(i - (long)n * K);
    dst[i] = src[(long)(rowOff + k) * 128 + n];
  }
}

// ---------------- per-node in-degree ---------------------------------------
__global__ void pna_count(const int* __restrict__ ei, float* __restrict__ cnt,
                          int E) {
  int e = blockIdx.x * blockDim.x + threadIdx.x;
  if (e < E) atom_add(&cnt[ei[(long)E + e]], 1.0f);
}

// ---------------- avg_log = mean(log(cnt+1)) accumulator -------------------
__global__ void pna_avglog(const float* __restrict__ cnt,
                           float* __restrict__ avga, int n) {
  __shared__ float red[256];
  int i = blockIdx.x * 256 + threadIdx.x;
  red[threadIdx.x] = (i < n) ? __builtin_logf(cnt[i] + 1.0f) : 0.f;
  __syncthreads();
  for (int s = 128; s > 0; s >>= 1) {
    if ((int)threadIdx.x < s) red[threadIdx.x] += red[threadIdx.x + s];
    __syncthreads();
  }
  if (threadIdx.x == 0) atom_add(avga, red[0]);
}

// ---------------- edge messages (WMMA) + atomic multi-aggregation ----------
// 128 edges/block, 8 waves; wave w owns cols [16w,16w+16), 8 row tiles.
// m = x[dst]@Wpre[0:128] + x[src]@Wpre[128:256] + edge_attr@Wc + bc
#define XS 132   // padded LDS row stride (132 % 64 == 4 -> no bank conflicts)
#define EPB 128  // edges per block
__global__ __launch_bounds__(256, 1)
void pna_edge(const float* __restrict__ x, const float* __restrict__ edge_attr,
              const float* __restrict__ WtD, const float* __restrict__ WtS,
              const float* __restrict__ Wct, const float* __restrict__ bc,
              const int* __restrict__ ei, float* __restrict__ sum,
              float* __restrict__ sumsq, float* __restrict__ gmx,
              float* __restrict__ gmn, int E) {
  __shared__ __align__(16) float sXd[EPB * XS];
  __shared__ __align__(16) float sXs[EPB * XS];
  __shared__ __align__(16) float sEA[EPB * 12];
  __shared__ int sSrc[EPB], sDst[EPB];

  const int tid = threadIdx.x;
  const int eb = blockIdx.x * EPB;

  if (tid < EPB) {
    int e = eb + tid;
    sSrc[tid] = (e < E) ? ei[e] : 0;
    sDst[tid] = (e < E) ? ei[(long)E + e] : 0;
  }
  __syncthreads();

  {  // gather x rows: 2 threads per row, 16 float4 each
    int r = tid >> 1, g = tid & 1;
    const f4* xs = (const f4*)(x + (long)sSrc[r] * 128);
    const f4* xd = (const f4*)(x + (long)sDst[r] * 128);
    f4* ds = (f4*)&sXs[r * XS];
    f4* dd = (f4*)&sXd[r * XS];
#pragma unroll
    for (int j = 0; j < 16; ++j) {
      int c = g + 2 * j;
      ds[c] = xs[c];
      dd[c] = xd[c];
    }
  }
  for (int i = tid; i < EPB * 12; i += 256) {  // edge attrs, zero-padded to 12
    int r = i / 12, c = i - r * 12;
    int e = eb + r;
    sEA[i] = (c < 10 && e < E) ? edge_attr[(long)e * 10 + c] : 0.f;
  }
  __syncthreads();

  const int lane = tid & 31;
  const int wave = tid >> 5;
  const int nloc = lane & 15;
  const int n = wave * 16 + nloc;        // output column
  const int kh = (lane >> 4) << 1;       // K offset 0 or 2 (A/B frag split)
  const int rh = (lane >> 4) << 3;       // C row offset 0 or 8

  const float* wd = WtD + n * 128;       // transposed: pair contiguous
  const float* wsr = WtS + n * 128;
  const float* wc = Wct + n * 12;

  v8f acc[8] = {};

  for (int k0 = 0; k0 < 128; k0 += 4) {  // x[dst] block
    int k = k0 + kh;
    v2f b = *(const v2f*)(wd + k);
#pragma unroll
    for (int rt = 0; rt < 8; ++rt) {
      v2f a = *(const v2f*)&sXd[(rt * 16 + nloc) * XS + k];
      acc[rt] = wmma_f32(a, b, acc[rt]);
    }
  }
  for (int k0 = 0; k0 < 128; k0 += 4) {  // x[src] block
    int k = k0 + kh;
    v2f b = *(const v2f*)(wsr + k);
#pragma unroll
    for (int rt = 0; rt < 8; ++rt) {
      v2f a = *(const v2f*)&sXs[(rt * 16 + nloc) * XS + k];
      acc[rt] = wmma_f32(a, b, acc[rt]);
    }
  }
#pragma unroll
  for (int k0 = 0; k0 < 12; k0 += 4) {   // folded edge-encoder block, K=12
    int k = k0 + kh;
    v2f b = *(const v2f*)(wc + k);
#pragma unroll
    for (int rt = 0; rt < 8; ++rt) {
      v2f a = *(const v2f*)&sEA[(rt * 16 + nloc) * 12 + k];
      acc[rt] = wmma_f32(a, b, acc[rt]);
    }
  }

  const float bias = bc[n];
#pragma unroll
  for (int rt = 0; rt < 8; ++rt) {
#pragma unroll
    for (int r = 0; r < 8; ++r) {
      int row = rt * 16 + r + rh;        // edge within block tile
      int e = eb + row;
      if (e < E) {
        float v = acc[rt][r] + bias;
        long o = (long)sDst[row] * 128 + n;
        atom_add(&sum[o], v);
        atom_add(&sumsq[o], v * v);
        atom_max(&gmx[o], v);
        atom_min(&gmn[o], v);
      }
    }
  }
}

// ---------------- node post-MLP + final linear (WMMA, fused) ---------------
// out = (x@Wpx + agg@Wp1 + a*(agg@Wp2) + b*(agg@Wp3) + b_post) @ W_lin + b_lin
// WtP is W_post transposed: WtP[n*1664 + k]; WtL is W_lin transposed.
#define AS 516   // padded agg row stride
__global__ __launch_bounds__(256, 1)
void pna_node(const float* __restrict__ x, const float* __restrict__ WtP,
              const float* __restrict__ b_post, const float* __restrict__ WtL,
              const float* __restrict__ b_lin, const float* __restrict__ sum,
              const float* __restrict__ sumsq, const float* __restrict__ gmx,
              const float* __restrict__ gmn, const float* __restrict__ cnt,
              const float* __restrict__ avga, float* __restrict__ out,
              int Nn) {
  __shared__ __align__(16) float sX[32 * XS];
  __shared__ __align__(16) float sAgg[32 * AS];  // reused as sMid later
  __shared__ float sAlpha[32], sBeta[32];

  const int tid = threadIdx.x;
  const int nb = blockIdx.x * 32;
  const float avg_log = avga[0] / (float)Nn;

  {  // build [mean,max,min,std] agg rows + scalers; 8 threads per node
    int r = tid >> 3, g = tid & 7;
    int node = nb + r;
    bool ok = node < Nn;
    float c = ok ? cnt[node] : 0.f;
    float deg = fmaxf(c, 1.0f);
    bool has = c > 0.f;
    if (g == 0) {
      float slog = __builtin_logf(deg + 1.0f);
      sAlpha[r] = slog / avg_log;
      sBeta[r] = avg_log / slog;
    }
    long base = (long)(ok ? node : 0) * 128;
    float* ar = &sAgg[r * AS];
    for (int j = 0; j < 16; ++j) {
      int col = g * 16 + j;
      float s = ok ? sum[base + col] : 0.f;
      float sq = ok ? sumsq[base + col] : 0.f;
      float mean = s / deg;
      float var = sq / deg - mean * mean;
      var = var > 0.f ? var : 0.f;
      ar[col] = mean;
      ar[128 + col] = (has && ok) ? gmx[base + col] : 0.f;
      ar[256 + col] = (has && ok) ? gmn[base + col] : 0.f;
      ar[384 + col] = __builtin_sqrtf(var + EPS_PNA);
      sX[r * XS + col] = ok ? x[base + col] : 0.f;
    }
  }
  __syncthreads();

  const int lane = tid & 31;
  const int wave = tid >> 5;
  const int nloc = lane & 15;
  const int n = wave * 16 + nloc;
  const int kh = (lane >> 4) << 1;
  const int rh = (lane >> 4) << 3;

  const float* wp = WtP + (long)n * 1664;  // one row of W_post^T
  const float* wl = WtL + n * 128;

  v8f ax[2] = {};
  v8f y1[2] = {};
  v8f y2[2] = {};
  v8f y3[2] = {};

  for (int k0 = 0; k0 < 128; k0 += 4) {  // x @ W_post[0:128]
    int k = k0 + kh;
    v2f b = *(const v2f*)(wp + k);
#pragma unroll
    for (int rt = 0; rt < 2; ++rt) {
      v2f a = *(const v2f*)&sX[(rt * 16 + nloc) * XS + k];
      ax[rt] = wmma_f32(a, b, ax[rt]);
    }
  }
  for (int k0 = 0; k0 < 512; k0 += 4) {  // shared A-frag, 3 B streams
    int k = k0 + kh;
    v2f b1 = *(const v2f*)(wp + 128 + k);   // identity-scaler block
    v2f b2 = *(const v2f*)(wp + 640 + k);   // amplification block
    v2f b3 = *(const v2f*)(wp + 1152 + k);  // attenuation block
#pragma unroll
    for (int rt = 0; rt < 2; ++rt) {
      v2f a = *(const v2f*)&sAgg[(rt * 16 + nloc) * AS + k];
      y1[rt] = wmma_f32(a, b1, y1[rt]);
      y2[rt] = wmma_f32(a, b2, y2[rt]);
      y3[rt] = wmma_f32(a, b3, y3[rt]);
    }
  }
  __syncthreads();  // done reading sAgg everywhere; reuse as sMid
  float* sMid = sAgg;
  const float bp = b_post[n];
#pragma unroll
  for (int rt = 0; rt < 2; ++rt) {
#pragma unroll
    for (int r = 0; r < 8; ++r) {
      int row = rt * 16 + r + rh;
      float v = ax[rt][r] + y1[rt][r] + sAlpha[row] * y2[rt][r] +
                sBeta[row] * y3[rt][r] + bp;
      sMid[row * XS + n] = v;
    }
  }
  __syncthreads();

  v8f o[2] = {};
  for (int k0 = 0; k0 < 128; k0 += 4) {  // mid @ W_lin
    int k = k0 + kh;
    v2f b = *(const v2f*)(wl + k);
#pragma unroll
    for (int rt = 0; rt < 2; ++rt) {
      v2f a = *(const v2f*)&sMid[(rt * 16 + nloc) * XS + k];
      o[rt] = wmma_f32(a, b, o[rt]);
    }
  }
  const float bl = b_lin[n];
#pragma unroll
  for (int rt = 0; rt < 2; ++rt) {
#pragma unroll
    for (int r = 0; r < 8; ++r) {
      int row = rt * 16 + r + rh;
      int node = nb + row;
      if (node < Nn) out[(long)node * 128 + n] = o[rt][r] + bl;
    }
  }
}

extern "C" void kernel_launch(void* const* d_in, const int* in_sizes, int n_in,
                              void* d_out, int out_size, void* d_ws,
                              size_t ws_size, hipStream_t stream) {
  const float* x = (const float*)d_in[0];
  const float* edge_attr = (const float*)d_in[1];
  const float* W_ee = (const float*)d_in[2];
  const float* b_ee = (const float*)d_in[3];
  const float* W_pre = (const float*)d_in[4];
  const float* b_pre = (const float*)d_in[5];
  const float* W_post = (const float*)d_in[6];
  const float* b_post = (const float*)d_in[7];
  const float* W_lin = (const float*)d_in[8];
  const float* b_lin = (const float*)d_in[9];
  const int* ei = (const int*)d_in[10];
  float* out = (float*)d_out;

  const int N = in_sizes[0] / 128;
  const int E = in_sizes[1] / 10;

  // ws layout (floats):
  // sum | sumsq | max | min | cnt | avga | Wct | bc | WtD | WtS | WtP | WtL
  float* w = (float*)d_ws;
  long ND = (long)N * 128;
  float* sum = w;
  float* sumsq = w + ND;
  float* gmx = w + 2 * ND;
  float* gmn = w + 3 * ND;
  float* cnt = w + 4 * ND;
  float* avga = cnt + N;
  float* Wct = avga + 1;          // 128*12
  float* bc = Wct + 128 * 12;     // 128
  float* WtD = bc + 128;          // 128*128
  float* WtS = WtD + 128 * 128;   // 128*128
  float* WtP = WtS + 128 * 128;   // 128*1664
  float* WtL = WtP + 128 * 1664;  // 128*128

  long T = ND;
  pna_init<<<(unsigned)((T + 255) / 256), 256, 0, stream>>>(sum, sumsq, gmx,
                                                            gmn, cnt, avga, T,
                                                            N);
  pna_prep<<<1, 128, 0, stream>>>(W_ee, b_ee, W_pre, b_pre, Wct, bc);
  pna_transpose<<<(128 * 128 + 255) / 256, 256, 0, stream>>>(W_pre, WtD, 128,
                                                             0);
  pna_transpose<<<(128 * 128 + 255) / 256, 256, 0, stream>>>(W_pre, WtS, 128,
                                                             128);
  pna_transpose<<<(128 * 1664 + 255) / 256, 256, 0, stream>>>(W_post, WtP,
                                                              1664, 0);
  pna_transpose<<<(128 * 128 + 255) / 256, 256, 0, stream>>>(W_lin, WtL, 128,
                                                             0);
  pna_count<<<(E + 255) / 256, 256, 0, stream>>>(ei, cnt, E);
  pna_edge<<<(E + EPB - 1) / EPB, 256, 0, stream>>>(x, edge_attr, WtD, WtS,
                                                    Wct, bc, ei, sum, sumsq,
                                                    gmx, gmn, E);
  pna_avglog<<<(N + 255) / 256, 256, 0, stream>>>(cnt, avga, N);
  pna_node<<<(N + 31) / 32, 256, 0, stream>>>(x, WtP, b_post, WtL, b_lin, sum,
                                              sumsq, gmx, gmn, cnt, avga, out,
                                              N);
}